// HyperGraphAttention_88055419503319
// MI455X (gfx1250) — compile-verified
//
#include <hip/hip_runtime.h>
#include <hip/hip_bf16.h>

#define N_NODES 50000
#define N_EDGES 800000
#define IN_F    128
#define OUT_F   64
#define M_TILES (N_NODES / 16)   // 3125 exactly

typedef __attribute__((ext_vector_type(2))) float v2f;
typedef __attribute__((ext_vector_type(8))) float v8f;

// ---------------------------------------------------------------------------
// Kernel 0: out[i][f] = bias[f]  (bias folded into accumulator), rowsum = 0
// ---------------------------------------------------------------------------
__global__ void gat_init(float* __restrict__ out, const float* __restrict__ bias,
                         float* __restrict__ rowsum) {
    int i = blockIdx.x * blockDim.x + threadIdx.x;
    if (i < N_NODES * OUT_F) out[i] = bias[i & (OUT_F - 1)];
    if (i < N_NODES) rowsum[i] = 0.0f;
}

// ---------------------------------------------------------------------------
// Kernel 1: H' = H x W via V_WMMA_F32_16X16X4_F32 (full fp32 precision).
// One wave computes a 16x64 tile: 4 accumulator fragments, 32 k-steps of 4.
// ---------------------------------------------------------------------------
__global__ void __launch_bounds__(256)
gat_gemm_wmma(const float* __restrict__ H, const float* __restrict__ W,
              float* __restrict__ Hp) {
    const int lane = threadIdx.x & 31;
    const int wave = threadIdx.x >> 5;
    const int tile = blockIdx.x * 8 + wave;   // wave-uniform
    if (tile >= M_TILES) return;              // whole wave exits together (EXEC all-1s kept)

    const int row0  = tile * 16;
    const int m     = lane & 15;              // M index (and N index for B/C)
    const int khalf = (lane >> 4) * 2;        // lanes 16-31 handle K+2,K+3

    const float* __restrict__ arow = H + (size_t)(row0 + m) * IN_F;

    v8f c0 = {}, c1 = {}, c2 = {}, c3 = {};

    #pragma unroll 4
    for (int k = 0; k < IN_F; k += 4) {
        // A fragment: 16x4, lane holds H[row0+m][k+khalf], H[row0+m][k+khalf+1]
        v2f a;
        a.x = arow[k + khalf + 0];
        a.y = arow[k + khalf + 1];

        // B fragments: 4x16 per N-tile; lane holds W[k+khalf][n], W[k+khalf+1][n]
        const float* __restrict__ wr = W + (size_t)(k + khalf) * OUT_F;
        v2f b0, b1, b2, b3;
        b0.x = wr[ 0 + m]; b0.y = wr[OUT_F +  0 + m];
        b1.x = wr[16 + m]; b1.y = wr[OUT_F + 16 + m];
        b2.x = wr[32 + m]; b2.y = wr[OUT_F + 32 + m];
        b3.x = wr[48 + m]; b3.y = wr[OUT_F + 48 + m];

        c0 = __builtin_amdgcn_wmma_f32_16x16x4_f32(false, a, false, b0, (short)0, c0, false, false);
        c1 = __builtin_amdgcn_wmma_f32_16x16x4_f32(false, a, false, b1, (short)0, c1, false, false);
        c2 = __builtin_amdgcn_wmma_f32_16x16x4_f32(false, a, false, b2, (short)0, c2, false, false);
        c3 = __builtin_amdgcn_wmma_f32_16x16x4_f32(false, a, false, b3, (short)0, c3, false, false);
    }

    // C/D layout: VGPR g, lanes 0-15 -> M=g, lanes 16-31 -> M=g+8; N = lane&15
    const int mofs = 8 * (lane >> 4);
    float* __restrict__ dst = Hp + (size_t)(row0 + mofs) * OUT_F + m;
    #pragma unroll
    for (int g = 0; g < 8; ++g) {
        dst[(size_t)g * OUT_F +  0] = c0[g];
        dst[(size_t)g * OUT_F + 16] = c1[g];
        dst[(size_t)g * OUT_F + 32] = c2[g];
        dst[(size_t)g * OUT_F + 48] = c3[g];
    }
}

// ---------------------------------------------------------------------------
// Kernel 2: per-edge score = leaky_relu(<H'[r], H'[c]>), exp, row segment sum.
// One wave per edge; lane l handles features 2l, 2l+1 (coalesced 128B lines).
// ---------------------------------------------------------------------------
__global__ void __launch_bounds__(256)
gat_scores(const float* __restrict__ Hp, const int* __restrict__ erow,
           const int* __restrict__ ecol, float* __restrict__ exp_s,
           float* __restrict__ rowsum) {
    const int e    = (blockIdx.x * blockDim.x + threadIdx.x) >> 5;
    const int lane = threadIdx.x & 31;
    if (e >= N_EDGES) return;

    const int r = erow[e];
    const int c = ecol[e];

    const float2 a = ((const float2*)(Hp + (size_t)r * OUT_F))[lane];
    const float2 b = ((const float2*)(Hp + (size_t)c * OUT_F))[lane];
    float p = a.x * b.x + a.y * b.y;

    #pragma unroll
    for (int off = 16; off > 0; off >>= 1)
        p += __shfl_xor(p, off, 32);

    if (lane == 0) {
        const float s  = (p > 0.0f) ? p : 0.2f * p;   // leaky_relu, slope 0.2
        const float es = expf(s);
        exp_s[e] = es;
        atomicAdd(rowsum + r, es);                    // global_atomic_add_f32
    }
}

// ---------------------------------------------------------------------------
// Kernel 3: alpha = exp_s / (rowsum[r] + 1e-10); out[r] += alpha * H'[c]
// ---------------------------------------------------------------------------
__global__ void __launch_bounds__(256)
gat_scatter(const float* __restrict__ Hp, const int* __restrict__ erow,
            const int* __restrict__ ecol, const float* __restrict__ exp_s,
            const float* __restrict__ rowsum, float* __restrict__ out) {
    const int e    = (blockIdx.x * blockDim.x + threadIdx.x) >> 5;
    const int lane = threadIdx.x & 31;
    if (e >= N_EDGES) return;

    const int r = erow[e];
    const int c = ecol[e];
    const float alpha = exp_s[e] / (rowsum[r] + 1e-10f);

    const float2 hc = ((const float2*)(Hp + (size_t)c * OUT_F))[lane];
    float* __restrict__ dst = out + (size_t)r * OUT_F + 2 * lane;
    atomicAdd(dst + 0, alpha * hc.x);
    atomicAdd(dst + 1, alpha * hc.y);
}

// ---------------------------------------------------------------------------
extern "C" void kernel_launch(void* const* d_in, const int* in_sizes, int n_in,
                              void* d_out, int out_size, void* d_ws, size_t ws_size,
                              hipStream_t stream) {
    const float* H    = (const float*)d_in[0];   // [50000,128]
    const float* W    = (const float*)d_in[1];   // [128,64]
    const float* bias = (const float*)d_in[2];   // [64]
    const int*   erow = (const int*)d_in[3];     // [800000]
    const int*   ecol = (const int*)d_in[4];     // [800000]
    float* out = (float*)d_out;                  // [50000,64]

    float* Hp     = (float*)d_ws;                // 50000*64 f32
    float* exp_s  = Hp + (size_t)N_NODES * OUT_F;// 800000 f32
    float* rowsum = exp_s + N_EDGES;             // 50000 f32

    // 0) out = bias (broadcast), rowsum = 0
    gat_init<<<(N_NODES * OUT_F + 255) / 256, 256, 0, stream>>>(out, bias, rowsum);

    // 1) H' = H x W  (WMMA f32 16x16x4; 3125 row-tiles, 8 waves/block)
    gat_gemm_wmma<<<(M_TILES + 7) / 8, 256, 0, stream>>>(H, W, Hp);

    // 2) edge scores -> exp_s, rowsum (wave per edge: 8 edges per 256-thread block)
    gat_scores<<<N_EDGES / 8, 256, 0, stream>>>(Hp, erow, ecol, exp_s, rowsum);

    // 3) normalize + SpMM scatter + (bias already in out)
    gat_scatter<<<N_EDGES / 8, 256, 0, stream>>>(Hp, erow, ecol, exp_s, rowsum, out);
}